// Model_22797686408015
// MI455X (gfx1250) — compile-verified
//
#include <hip/hip_runtime.h>
#include <hip/hip_bf16.h>

// Streaming clamp: out = min(max(x, lo), hi)
//
// Roofline: 33.55M f32 -> 256 MiB HBM traffic, ~67 MFLOP => pure bandwidth,
// floor ~11.5 us at 23.3 TB/s. Strategy:
//   - branch-free main kernel, grid covers exactly (n/4096)*4096 elements
//   - s_clause'd 4x global_load_b128 th:TH_LOAD_NT per thread (lane-contiguous)
//   - v_med3_f32 single-instruction clamp (no NaN-canonicalize preamble)
//   - non-temporal loads AND stores (touch-once data, keep L2 clean)
//   - scalar tail kernel only if n % 4096 != 0 (not for reference shape)

typedef __attribute__((ext_vector_type(4))) float v4f;

__device__ __forceinline__ v4f clamp4(v4f a, float lo, float hi) {
    // v_med3_f32: median(x, lo, hi) == clamp(x, lo, hi) for lo <= hi.
    a.x = __builtin_amdgcn_fmed3f(a.x, lo, hi);
    a.y = __builtin_amdgcn_fmed3f(a.y, lo, hi);
    a.z = __builtin_amdgcn_fmed3f(a.z, lo, hi);
    a.w = __builtin_amdgcn_fmed3f(a.w, lo, hi);
    return a;
}

// Processes exactly gridDim.x * 4096 floats. No bounds checks, no branches:
// hot path is clause(4x b128 NT load) -> 16x v_med3 -> 4x b128 NT store.
__global__ __launch_bounds__(256) void clamp_main(
    const v4f* __restrict__ x,
    const float* __restrict__ clamp_params,
    v4f* __restrict__ out)
{
    const float lo = clamp_params[0];   // uniform s_load broadcast
    const float hi = clamp_params[1];

    // Block covers 1024 consecutive float4s; each of the 4 loads is
    // lane-contiguous (32 lanes x 16B = 512B per load instruction).
    const int base = blockIdx.x * 1024 + threadIdx.x;

    v4f a0 = __builtin_nontemporal_load(&x[base]);
    v4f a1 = __builtin_nontemporal_load(&x[base + 256]);
    v4f a2 = __builtin_nontemporal_load(&x[base + 512]);
    v4f a3 = __builtin_nontemporal_load(&x[base + 768]);

    a0 = clamp4(a0, lo, hi);
    a1 = clamp4(a1, lo, hi);
    a2 = clamp4(a2, lo, hi);
    a3 = clamp4(a3, lo, hi);

    __builtin_nontemporal_store(a0, &out[base]);
    __builtin_nontemporal_store(a1, &out[base + 256]);
    __builtin_nontemporal_store(a2, &out[base + 512]);
    __builtin_nontemporal_store(a3, &out[base + 768]);
}

// Scalar remainder kernel (general-shape safety; not launched for the
// reference shape since 33,554,432 % 4096 == 0).
__global__ void clamp_tail(
    const float* __restrict__ x,
    const float* __restrict__ clamp_params,
    float* __restrict__ out,
    int start, int n)
{
    const float lo = clamp_params[0];
    const float hi = clamp_params[1];
    int i = start + blockIdx.x * blockDim.x + threadIdx.x;
    if (i < n) {
        out[i] = __builtin_amdgcn_fmed3f(x[i], lo, hi);
    }
}

extern "C" void kernel_launch(void* const* d_in, const int* in_sizes, int n_in,
                              void* d_out, int out_size, void* d_ws, size_t ws_size,
                              hipStream_t stream) {
    const float* x  = (const float*)d_in[0];
    const float* cp = (const float*)d_in[1];
    float* out      = (float*)d_out;

    const int n = out_size;                 // 33,554,432 for reference
    const int ELEMS_PER_BLOCK = 4096;       // 256 threads * 4 float4 * 4 floats
    const int fullBlocks = n / ELEMS_PER_BLOCK;

    if (fullBlocks > 0) {
        clamp_main<<<fullBlocks, 256, 0, stream>>>(
            (const v4f*)x, cp, (v4f*)out);
    }

    const int done = fullBlocks * ELEMS_PER_BLOCK;
    if (done < n) {
        const int rem = n - done;
        const int block = 256;
        const int grid = (rem + block - 1) / block;
        clamp_tail<<<grid, block, 0, stream>>>(x, cp, out, done, n);
    }
}